// myLoss_20538533609577
// MI455X (gfx1250) — compile-verified
//
#include <hip/hip_runtime.h>

// Problem dims
#define TDIM 200
#define VDIM 5000
#define BDIM 1024
#define TP   208    // 13 * 16
#define VP   5008   // 313 * 16
#define NTT  13     // T tiles of 16
#define NVT  313    // V tiles of 16
#define NK1  50     // k-chunks in K^T@u contraction (t < 200)
#define NK2  52     // t-tile*k-chunk slots in K@w (13 tiles * 4 chunks)
#define PF   8      // A-operand prefetch depth (in-flight global_load_b64)

typedef __attribute__((ext_vector_type(2))) float v2f;
typedef __attribute__((ext_vector_type(8))) float v8f;

// Explicit global address space -> global_load_* (LOADcnt only)
typedef __attribute__((address_space(1))) const float* cgp;
typedef __attribute__((address_space(1))) const v2f*  cgv2;
#define AS_G(p)  ((cgp)(unsigned long long)(p))

// D = A(16x4,f32) * B(4x16,f32) + C(16x16,f32)  -- V_WMMA_F32_16X16X4_F32
__device__ __forceinline__ v8f wmma4(v2f a, v2f b, v8f c) {
  return __builtin_amdgcn_wmma_f32_16x16x4_f32(
      /*neg_a=*/false, a, /*neg_b=*/false, b,
      /*c_mod=*/(short)0, c, /*reuse_a=*/false, /*reuse_b=*/false);
}

// pair-interleaved LDS index for (t,c): rows t,t+1 adjacent per column
#define UIDX(t, c) ((((t) >> 1) * 32) + ((c) * 2) + ((t) & 1))

// ---------------------------------------------------------------------------
// Prep. Builds WMMA A-operand-order ("op") images of K = exp(-alpha*M):
//  K1op : A-chunks for t = K^T @ u   (m = v, k = t), 50 chunks/vtile
//  K2op : A-chunks for s = K  @ w    (m = t, k = v), 52 slots/vtile
//  Km2op: same as K2op but for K*M (final divergence pass)
// Chunk element order == lane order: (m,kc) at lane*2+slot,
// lane = m + 16*(kc>>1), slot = kc&1  -> coalesced global_load_b64.
// ---------------------------------------------------------------------------
__global__ __launch_bounds__(256) void prep_kernel(
    const float* __restrict__ M, const float* __restrict__ dt,
    const float* __restrict__ dw, const float* __restrict__ alpha_p,
    float* __restrict__ K1op, float* __restrict__ K2op,
    float* __restrict__ Km2op, float* __restrict__ aT,
    float* __restrict__ aTpair, float* __restrict__ bT,
    float* __restrict__ accum) {
  const float alpha = alpha_p[0];
  const long nK = (long)TP * VP;
  const long nA = (long)TP * BDIM;
  const long nB = (long)VP * BDIM;
  const long total = nK + nA + nB;
  for (long i = (long)blockIdx.x * 256 + threadIdx.x; i < total;
       i += (long)gridDim.x * 256) {
    if (i < nK) {
      int t = (int)(i / VP), v = (int)(i % VP);
      float kv = 0.f, km = 0.f;
      if (t < TDIM && v < VDIM) {
        float mv = M[t * VDIM + v];
        kv = __expf(-alpha * mv);
        km = kv * mv;
      }
      int vt = v >> 4, mm = v & 15;
      if (t < TDIM) {  // K1op: m = v-in-tile, k = t
        int k4 = t >> 2, kc = t & 3;
        unsigned lane = mm + ((kc >> 1) << 4);
        K1op[((unsigned)(vt * NK1 + k4) << 6) + (lane << 1) + (kc & 1)] = kv;
      }
      {  // K2op/Km2op: m = t-in-tile, k = v-in-tile
        int tt = t >> 4, tm = t & 15;
        int k4 = mm >> 2, kc = mm & 3;
        unsigned lane = tm + ((kc >> 1) << 4);
        unsigned idx =
            ((unsigned)(vt * NK2 + tt * 4 + k4) << 6) + (lane << 1) + (kc & 1);
        K2op[idx] = kv;
        Km2op[idx] = km;
      }
    } else if (i < nK + nA) {
      long j = i - nK;
      int t = (int)(j / BDIM), c = (int)(j % BDIM);
      float av = (t < TDIM) ? dt[c * TDIM + t] : 0.f;
      aT[j] = av;
      aTpair[(((long)(t >> 1) * BDIM + c) << 1) + (t & 1)] = av;
    } else {
      long j = i - nK - nA;
      int v = (int)(j / BDIM), c = (int)(j % BDIM);
      bT[j] = (v < VDIM) ? dw[c * VDIM + v] : 0.f;
    }
  }
  if (blockIdx.x == 0 && threadIdx.x < 8) accum[threadIdx.x] = 0.f;
}

// ---------------------------------------------------------------------------
// Sinkhorn: each block owns 16 batch columns; full while-loop in one launch.
// ---------------------------------------------------------------------------
__global__ __launch_bounds__(256) void sinkhorn_kernel(
    const float* __restrict__ K1op_, const float* __restrict__ K2op_,
    const float* __restrict__ Km2op_, const float* __restrict__ aT_,
    const float* __restrict__ bT_, float* __restrict__ accum) {
  cgp K1op  = AS_G(K1op_);
  cgp K2op  = AS_G(K2op_);
  cgp Km2op = AS_G(Km2op_);
  cgp aT    = AS_G(aT_);
  cgp bT    = AS_G(bT_);

  __shared__ __align__(16) float u_lds[TP * 16];   // pair-interleaved
  __shared__ __align__(16) float up_lds[TP * 16];  // pair-interleaved
  __shared__ float s_lds[TP * 16];                 // linear [t*16+c]
  __shared__ __align__(16) float w_lds[8][256];    // per-wave, pair-interleaved
  __shared__ float err_lds[16];
  __shared__ float red_lds[16];
  __shared__ int convflag;

  const int tid = threadIdx.x;
  const int wid = tid >> 5;
  const int lane = tid & 31;
  const unsigned m = lane & 15;                 // tile col / A-row
  const unsigned rbase = (lane < 16) ? 0u : 8u; // D-row base for this half
  const unsigned j0 = blockIdx.x * 16;
  const unsigned lpair = (unsigned)lane * 2;    // lane base in operand chunks

  // t_tile = (K^T @ u) with an explicit 8-deep rotating A-prefetch:
  // distinct destination regs => staggered s_wait_loadcnt, ~PF loads in flight
  auto ktu_tile = [&](const float* ub, unsigned vt) -> v8f {
    v8f acc = {};
    const unsigned o1 = vt * (NK1 * 64u) + lpair;
    v2f apf[PF];
#pragma unroll
    for (int p = 0; p < PF; ++p)
      apf[p] = *(cgv2)&K1op[o1 + (unsigned)(p * 64)];
#pragma unroll
    for (int k4 = 0; k4 < NK1; ++k4) {
      v2f a = apf[k4 & (PF - 1)];
      if (k4 + PF < NK1)
        apf[k4 & (PF - 1)] = *(cgv2)&K1op[o1 + (unsigned)((k4 + PF) * 64)];
      v2f b = *(const v2f*)&ub[lpair + (unsigned)(k4 * 32)];
      acc = wmma4(a, b, acc);
    }
    return acc;
  };

  // s(+)= K2 @ (bT / (K^T @ u)) over this wave's V chunks
  auto sweep = [&](cgp K2) {
    float* wl = w_lds[wid];
    v8f accS[NTT];
#pragma unroll
    for (int t = 0; t < NTT; ++t) accS[t] = (v8f){};
    for (unsigned vt = wid; vt < NVT; vt += 8) {
      v8f tv = ktu_tile(u_lds, vt);
      const unsigned bTb = vt * (16u * BDIM) + rbase * BDIM + j0 + m;
#pragma unroll
      for (int r = 0; r < 8; ++r) {               // w = b / t (pad rows -> 0)
        unsigned vl = rbase + r;
        unsigned vg = vt * 16 + vl;
        float wv = 0.f;
        if (vg < VDIM) wv = bT[bTb + (unsigned)(r * BDIM)] / tv[r];
        wl[((vl >> 1) * 32) + m * 2 + (vl & 1)] = wv;
      }
      // all 4 w-chunks of this vtile into registers
      v2f b4[4];
#pragma unroll
      for (int k4 = 0; k4 < 4; ++k4)
        b4[k4] = *(const v2f*)&wl[lpair + (unsigned)(k4 * 32)];
      // flattened 52-slot stream (k4-major, tt-minor) with rotating prefetch
      const unsigned o2 = vt * (NK2 * 64u) + lpair;
      v2f apf[PF];
#pragma unroll
      for (int p = 0; p < PF; ++p) {
        int k4 = p / NTT, tt = p % NTT;
        apf[p] = *(cgv2)&K2[o2 + (unsigned)((tt * 4 + k4) * 64)];
      }
#pragma unroll
      for (int f = 0; f < NK2; ++f) {
        int k4 = f / NTT, tt = f % NTT;
        v2f a = apf[f & (PF - 1)];
        if (f + PF < NK2) {
          int k4n = (f + PF) / NTT, ttn = (f + PF) % NTT;
          apf[f & (PF - 1)] =
              *(cgv2)&K2[o2 + (unsigned)((ttn * 4 + k4n) * 64)];
        }
        accS[tt] = wmma4(a, b4[k4], accS[tt]);
      }
    }
#pragma unroll
    for (int t = 0; t < NTT; ++t)
#pragma unroll
      for (int r = 0; r < 8; ++r)
        atomicAdd(&s_lds[(t * 16 + rbase + r) * 16 + m], accS[t][r]);
  };

  auto body = [&]() {  // u = a / (K @ (b / (K^T u)))
    for (int idx = tid; idx < TP * 16; idx += 256) s_lds[idx] = 0.f;
    __syncthreads();
    sweep(K2op);
    __syncthreads();
    for (int idx = tid; idx < TP * 16; idx += 256) {
      int t = idx >> 4, c = idx & 15;
      float un = 0.f;
      if (t < TDIM) un = aT[(unsigned)t * BDIM + j0 + c] / s_lds[idx];
      u_lds[UIDX(t, c)] = un;
    }
    __syncthreads();
  };

  // init u = 1/T (pair layout)
  for (int idx = tid; idx < TP * 16; idx += 256) {
    int t = idx >> 4, c = idx & 15;
    u_lds[UIDX(t, c)] = (t < TDIM) ? (1.0f / TDIM) : 0.f;
  }
  __syncthreads();

  int cpt = 0;
  bool done = false;
  while (!done && cpt < 5000) {
    cpt++;
    body();
    if (cpt % 20 == 1) {  // refresh: extra update + marginal error check
      for (int idx = tid; idx < TP * 16; idx += 256) up_lds[idx] = u_lds[idx];
      __syncthreads();
      body();
      if (tid < 16) err_lds[tid] = 0.f;
      __syncthreads();
      float esum = 0.f;
      for (unsigned vt = wid; vt < NVT; vt += 8) {
        v8f to = ktu_tile(up_lds, vt);
        v8f tn = ktu_tile(u_lds, vt);
        const unsigned bTb = vt * (16u * BDIM) + rbase * BDIM + j0 + m;
#pragma unroll
        for (int r = 0; r < 8; ++r) {
          unsigned vg = vt * 16 + rbase + r;
          if (vg < VDIM) {
            float bval = bT[bTb + (unsigned)(r * BDIM)];
            float bb = (bval / to[r]) * tn[r];
            esum += fabsf(bb - bval);
          }
        }
      }
      atomicAdd(&err_lds[m], esum);
      __syncthreads();
      if (tid == 0) {
        float mx = 0.f;
        for (int c = 0; c < 16; ++c) mx = fmaxf(mx, err_lds[c]);
        convflag = (mx <= 0.005f) ? 1 : 0;
      }
      __syncthreads();
      done = (convflag != 0);
    }
  }

  // divergence: d_j = sum_t u[t][j] * ((K*M) @ (b/(K^T u)))[t][j]
  for (int idx = tid; idx < TP * 16; idx += 256) s_lds[idx] = 0.f;
  __syncthreads();
  sweep(Km2op);
  __syncthreads();
  if (tid < 16) {
    float d = 0.f;
    for (int t = 0; t < TDIM; ++t)
      d += u_lds[UIDX(t, tid)] * s_lds[t * 16 + tid];
    red_lds[tid] = d;
  }
  __syncthreads();
  if (tid == 0) {
    float tot = 0.f;
    for (int c = 0; c < 16; ++c) tot += red_lds[c];
    atomicAdd(&accum[0], tot);
  }
}

// ---------------------------------------------------------------------------
// Rec loss: logits = doc_topic @ (1-M) in [0,1] => log_softmax with fixed
// shift 1.0 (logits bounded) -> single sweep, no row-max pass.
// ---------------------------------------------------------------------------
__global__ __launch_bounds__(256) void rec_kernel(
    const float* __restrict__ aTpair_, const float* __restrict__ M_,
    const float* __restrict__ x_, float* __restrict__ accum) {
  cgp aTpair = AS_G(aTpair_);
  cgp Mg     = AS_G(M_);
  cgp xg     = AS_G(x_);

  __shared__ float se_s[16], sx_s[16], xs_s[16], red[16];
  const int tid = threadIdx.x;
  const int wid = tid >> 5;
  const int lane = tid & 31;
  const unsigned m = lane & 15;
  const unsigned kk = (lane < 16) ? 0u : 2u;
  const unsigned rbase = (lane < 16) ? 0u : 8u;
  const unsigned b0 = blockIdx.x * 16;

  if (tid < 16) { se_s[tid] = 0.f; sx_s[tid] = 0.f; xs_s[tid] = 0.f; }
  __syncthreads();

  float se[8], sxl[8], xsl[8];
#pragma unroll
  for (int r = 0; r < 8; ++r) { se[r] = 0.f; sxl[r] = 0.f; xsl[r] = 0.f; }

  const unsigned ab = ((b0 + m) << 1) + (kk >> 1) * (BDIM * 2);
  for (unsigned vt = wid; vt < NVT; vt += 8) {
    const unsigned v0 = vt * 16;
    const unsigned vg = v0 + m;
    const bool vok = (vg < VDIM);
    const unsigned mb = kk * VDIM + vg;
    v8f acc = {};
#pragma unroll
    for (int k = 0; k < TDIM; k += 4) {  // k+kk <= 199 always: no t guard
      v2f a = *(cgv2)&aTpair[ab + (unsigned)(k >> 1) * (BDIM * 2)];
      v2f b;
      b.x = vok ? (1.f - Mg[mb + (unsigned)k * VDIM]) : 0.f;
      b.y = vok ? (1.f - Mg[mb + (unsigned)(k + 1) * VDIM]) : 0.f;
      acc = wmma4(a, b, acc);
    }
    if (vok) {
#pragma unroll
      for (int r = 0; r < 8; ++r) {
        unsigned row = b0 + rbase + r;
        float lv = acc[r];
        float xv = xg[row * VDIM + vg];
        se[r] += __expf(lv - 1.0f);
        sxl[r] += xv * lv;
        xsl[r] += xv;
      }
    }
  }
#pragma unroll
  for (int r = 0; r < 8; ++r) {
    atomicAdd(&se_s[rbase + r], se[r]);
    atomicAdd(&sx_s[rbase + r], sxl[r]);
    atomicAdd(&xs_s[rbase + r], xsl[r]);
  }
  __syncthreads();
  if (tid < 16) {
    float lse = 1.0f + __logf(se_s[tid]);
    red[tid] = sx_s[tid] - lse * xs_s[tid];   // sum_v x*(logit - lse)
  }
  __syncthreads();
  if (tid == 0) {
    float tot = 0.f;
    for (int c = 0; c < 16; ++c) tot += red[c];
    atomicAdd(&accum[1], tot);
  }
}

__global__ void finalize_kernel(const float* __restrict__ accum,
                                const float* __restrict__ rw_p,
                                float* __restrict__ out) {
  if (blockIdx.x == 0 && threadIdx.x == 0) {
    float sh = accum[0] / (float)BDIM;
    float rec = -accum[1] / (float)BDIM;
    out[0] = rec;
    out[1] = sh;
    out[2] = rw_p[0] * rec + sh;
  }
}

// ---------------------------------------------------------------------------
extern "C" void kernel_launch(void* const* d_in, const int* in_sizes, int n_in,
                              void* d_out, int out_size, void* d_ws,
                              size_t ws_size, hipStream_t stream) {
  (void)in_sizes; (void)n_in; (void)out_size; (void)ws_size;
  const float* x   = (const float*)d_in[0];
  const float* dt  = (const float*)d_in[1];   // doc_topic [B,T]
  const float* dw  = (const float*)d_in[2];   // doc_word  [B,V]
  const float* M   = (const float*)d_in[3];   // [T,V]
  const float* alf = (const float*)d_in[5];   // sh_alpha
  const float* rw  = (const float*)d_in[6];   // rec_loss_weight
  float* out = (float*)d_out;

  float* K1op   = (float*)d_ws;                     // [NVT*50*64]
  float* K2op   = K1op + (size_t)NVT * NK1 * 64;    // [NVT*52*64]
  float* Km2op  = K2op + (size_t)NVT * NK2 * 64;    // [NVT*52*64]
  float* aT     = Km2op + (size_t)NVT * NK2 * 64;   // [TP,B]
  float* aTpair = aT + (size_t)TP * BDIM;           // [TP,B] pair-interleaved
  float* bT     = aTpair + (size_t)TP * BDIM;       // [VP,B]
  float* accum  = bT + (size_t)VP * BDIM;           // [8]

  prep_kernel<<<2048, 256, 0, stream>>>(M, dt, dw, alf, K1op, K2op, Km2op, aT,
                                        aTpair, bT, accum);
  sinkhorn_kernel<<<BDIM / 16, 256, 0, stream>>>(K1op, K2op, Km2op, aT, bT,
                                                 accum);
  rec_kernel<<<BDIM / 16, 256, 0, stream>>>(aTpair, M, x, accum);
  finalize_kernel<<<1, 64, 0, stream>>>(accum, rw, out);
}